// AttentionPooling_59880434041201
// MI455X (gfx1250) — compile-verified
//
#include <hip/hip_runtime.h>
#include <math.h>

typedef __attribute__((ext_vector_type(2))) float v2f;
typedef __attribute__((ext_vector_type(8))) float v8f;

#define NSEG 16384
#define DIMS 128
#define NEG_SLOPE 0.2f

// ---------------------------------------------------------------------------
// Pass 1: scores = LeakyReLU(x @ a) using V_WMMA_F32_16X16X4_F32.
// One wave handles 16 rows. A-tile = x[row_base+ (lane&15)][k + 2*(lane>>4) + j]
// (ISA 16x4 f32 A layout). B-tile broadcasts the attention-vector K-chunk
// across all 16 columns, so every column of the C tile equals the row's score.
// ---------------------------------------------------------------------------
__global__ void k_scores_wmma(const float* __restrict__ x,
                              const float* __restrict__ av,
                              float* __restrict__ scores,
                              int n_groups) {
  int wave = (int)((blockIdx.x * blockDim.x + threadIdx.x) >> 5);
  int lane = (int)(threadIdx.x & 31);
  if (wave >= n_groups) return;                 // uniform per-wave exit
  long row_base = (long)wave * 16;
  int m  = lane & 15;
  int hi = lane >> 4;                           // 0: lanes 0-15, 1: lanes 16-31
  const float* xrow = x + (row_base + m) * DIMS + 2 * hi;
  const float* arow = av + 2 * hi;
  v8f c = {};
#pragma unroll
  for (int k = 0; k < DIMS; k += 4) {
    v2f a = *(const v2f*)(xrow + k);            // A: VGPR0=K(2hi), VGPR1=K(2hi+1)
    v2f b = *(const v2f*)(arow + k);            // B: same K pairing, broadcast over N
    c = __builtin_amdgcn_wmma_f32_16x16x4_f32(
        /*neg_a=*/false, a, /*neg_b=*/false, b,
        /*c_mod=*/(short)0, c, /*reuse_a=*/false, /*reuse_b=*/false);
  }
  // C/D layout: lane L (N=L&15), VGPR v holds row M = v + 8*(L>>4); all N equal.
  if (m == 0) {
#pragma unroll
    for (int v = 0; v < 8; ++v) {
      float s = c[v];
      s = (s >= 0.f) ? s : NEG_SLOPE * s;       // LeakyReLU(0.2)
      scores[row_base + hi * 8 + v] = s;
    }
  }
}

// Tail rows (N % 16) via plain per-thread dot (not used for N=2M but kept general).
__global__ void k_scores_tail(const float* __restrict__ x,
                              const float* __restrict__ av,
                              float* __restrict__ scores,
                              long start, long n) {
  long i = start + blockIdx.x * blockDim.x + threadIdx.x;
  if (i >= n) return;
  float s = 0.f;
  for (int k = 0; k < DIMS; ++k) s = fmaf(x[i * DIMS + k], av[k], s);
  scores[i] = (s >= 0.f) ? s : NEG_SLOPE * s;
}

// ---------------------------------------------------------------------------
// Pass 2: segment start offsets via lower_bound on sorted int64 batch ids.
// seg_start has NSEG+1 entries; seg_start[NSEG] == n.
// ---------------------------------------------------------------------------
__global__ void k_offsets(const long long* __restrict__ batch,
                          int* __restrict__ seg_start, long n) {
  int s = (int)(blockIdx.x * blockDim.x + threadIdx.x);
  if (s > NSEG) return;
  long lo = 0, hi = n;
  while (lo < hi) {
    long mid = (lo + hi) >> 1;
    if (batch[mid] < (long long)s) lo = mid + 1; else hi = mid;
  }
  seg_start[s] = (int)lo;
}

// ---------------------------------------------------------------------------
// Pass 3: per-segment max and sum(exp(score - max)), one wave32 per segment.
// Fixed-order shuffle tree => deterministic. Empty segment: max guarded to 0.
// ---------------------------------------------------------------------------
__global__ void k_seg_stats(const float* __restrict__ scores,
                            const int* __restrict__ seg_start,
                            float* __restrict__ seg_max,
                            float* __restrict__ seg_sum) {
  int wave = (int)((blockIdx.x * blockDim.x + threadIdx.x) >> 5);
  int lane = (int)(threadIdx.x & 31);
  if (wave >= NSEG) return;
  int s0 = seg_start[wave], s1 = seg_start[wave + 1];

  float mx = -__builtin_inff();
  for (int i = s0 + lane; i < s1; i += 32) mx = fmaxf(mx, scores[i]);
#pragma unroll
  for (int off = 16; off > 0; off >>= 1) mx = fmaxf(mx, __shfl_xor(mx, off, 32));
  if (!__builtin_isfinite(mx)) mx = 0.f;        // reference's empty-segment guard

  float acc = 0.f;
  for (int i = s0 + lane; i < s1; i += 32) acc += expf(scores[i] - mx);
#pragma unroll
  for (int off = 16; off > 0; off >>= 1) acc += __shfl_xor(acc, off, 32);

  if (lane == 0) { seg_max[wave] = mx; seg_sum[wave] = acc; }
}

// ---------------------------------------------------------------------------
// Pass 4: out[s, col] = sum_i alpha_i * x[i, col]; one 128-thread block per
// segment, thread = column -> fully coalesced streaming of x; exactly one
// writer per output element (no pre-zero needed, empty segments write 0).
// ---------------------------------------------------------------------------
__global__ void k_pool(const float* __restrict__ x,
                       const float* __restrict__ scores,
                       const int* __restrict__ seg_start,
                       const float* __restrict__ seg_max,
                       const float* __restrict__ seg_sum,
                       float* __restrict__ out) {
  int s   = (int)blockIdx.x;
  int col = (int)threadIdx.x;                   // 0..127
  int s0 = seg_start[s], s1 = seg_start[s + 1];
  float mx  = seg_max[s];
  float inv = 1.f / (seg_sum[s] + 1e-16f);
  float acc = 0.f;
  for (int i = s0; i < s1; ++i) {
    float w = expf(scores[i] - mx) * inv;       // broadcast load of scores[i]
    acc = fmaf(w, x[(long)i * DIMS + col], acc);
  }
  out[(long)s * DIMS + col] = acc;
}

// ---------------------------------------------------------------------------
extern "C" void kernel_launch(void* const* d_in, const int* in_sizes, int n_in,
                              void* d_out, int out_size, void* d_ws, size_t ws_size,
                              hipStream_t stream) {
  const float*     x     = (const float*)d_in[0];
  const long long* batch = (const long long*)d_in[1];   // int64 segment ids
  const float*     av    = (const float*)d_in[2];
  float*           out   = (float*)d_out;

  long n = (long)in_sizes[1];                   // N rows (batch element count)

  // Workspace layout: scores[N] | seg_start[NSEG+1] | seg_max[NSEG] | seg_sum[NSEG]
  float* scores    = (float*)d_ws;
  int*   seg_start = (int*)(scores + n);
  float* seg_max   = (float*)(seg_start + NSEG + 1);
  float* seg_sum   = seg_max + NSEG;

  // Pass 1: WMMA GEMV scores
  long groups = n / 16;
  if (groups > 0) {
    long threads = groups * 32;
    int  block   = 256;
    unsigned grid = (unsigned)((threads + block - 1) / block);
    k_scores_wmma<<<grid, block, 0, stream>>>(x, av, scores, (int)groups);
  }
  long tail_start = groups * 16;
  if (tail_start < n) {
    k_scores_tail<<<1, 64, 0, stream>>>(x, av, scores, tail_start, n);
  }

  // Pass 2: segment offsets (binary search over sorted batch)
  k_offsets<<<(NSEG + 1 + 255) / 256, 256, 0, stream>>>(batch, seg_start, n);

  // Pass 3: segment max + sum(exp)
  k_seg_stats<<<(NSEG * 32 + 255) / 256, 256, 0, stream>>>(scores, seg_start,
                                                           seg_max, seg_sum);

  // Pass 4: attention-weighted pooling
  k_pool<<<NSEG, DIMS, 0, stream>>>(x, scores, seg_start, seg_max, seg_sum, out);
}